// E36LinearRecurrenceCell_79551384257214
// MI455X (gfx1250) — compile-verified
//
#include <hip/hip_runtime.h>
#include <math.h>

typedef __attribute__((ext_vector_type(2))) float    v2f;
typedef __attribute__((ext_vector_type(8))) float    v8f;
typedef __attribute__((ext_vector_type(4))) unsigned v4u;
typedef __attribute__((ext_vector_type(4))) int      v4i;
typedef __attribute__((ext_vector_type(8))) int      v8i;

#define TARGET_RADIUS 0.95f
#define SN_EPS 1e-8f

// -------------------------------------------------------------------------
// TDM: build a D# (groups 0/1 per CDNA5 ISA 8.3/8.4) and issue
// tensor_load_to_lds for a 2-D f32 tile [tile_y rows x tile_x cols],
// row stride = stride_x elements, destination packed at lds_off.
// 6-arg builtin form (clang-23 / therock-10.0 lane).
// -------------------------------------------------------------------------
__device__ __forceinline__ void tdm_load_2d_f32(unsigned lds_off,
                                                const void* gaddr,
                                                unsigned tile_x, unsigned tile_y,
                                                unsigned tensor_x, unsigned tensor_y,
                                                unsigned stride_x) {
  const unsigned long long ga = (unsigned long long)gaddr;
  v4u g0;
  g0.x = 1u;                                              // count=1, user D#
  g0.y = lds_off;                                         // lds_addr (bytes)
  g0.z = (unsigned)(ga & 0xFFFFFFFFu);                    // global_addr[31:0]
  g0.w = (unsigned)((ga >> 32) & 0x01FFFFFFu) | (2u << 30); // addr[56:32]|type=2
  v8i g1;
  g1[0] = (int)(2u << 16);                                // data_size=4B
  g1[1] = (int)((tensor_x & 0xFFFFu) << 16);              // tensor_dim0[15:0]
  g1[2] = (int)(((tensor_x >> 16) & 0xFFFFu) | ((tensor_y & 0xFFFFu) << 16));
  g1[3] = (int)(((tensor_y >> 16) & 0xFFFFu) | ((tile_x & 0xFFFFu) << 16));
  g1[4] = (int)(tile_y & 0xFFFFu);                        // tile_dim1, tile_dim2=0
  g1[5] = (int)stride_x;                                  // tensor_dim0_stride lo
  g1[6] = 0;
  g1[7] = 0;
  const v4i z4 = {0, 0, 0, 0};                            // groups 2/3 unused (2-D)
  const v8i z8 = {0, 0, 0, 0, 0, 0, 0, 0};
  __builtin_amdgcn_tensor_load_to_lds(g0, g1, z4, z4, z8, 0);
}

// -------------------------------------------------------------------------
// Block-wide sum reduction through LDS (blockDim.x power of two).
// -------------------------------------------------------------------------
__device__ __forceinline__ float block_sum(float v, float* sred) {
  const int tid = threadIdx.x;
  sred[tid] = v;
  __syncthreads();
  for (int s = blockDim.x >> 1; s > 0; s >>= 1) {
    if (tid < s) sred[tid] += sred[tid + s];
    __syncthreads();
  }
  float r = sred[0];
  __syncthreads();
  return r;
}

// -------------------------------------------------------------------------
// Kernel 1: spectral-norm power iteration (1 block, D threads).
// Also resets the grid-barrier counter each launch (graph-replay safe).
// -------------------------------------------------------------------------
__global__ void k_spectral(const float* __restrict__ Wh,
                           const float* __restrict__ u_in,
                           float* __restrict__ ws_scale,
                           unsigned* __restrict__ ctr,
                           int D) {
  __shared__ float su[1024];
  __shared__ float sv[1024];
  __shared__ float sred[1024];
  const int tid = threadIdx.x;

  su[tid] = u_in[tid];
  __syncthreads();

  for (int it = 0; it < 3; ++it) {
    float acc = 0.f;                                  // v = normalize(W^T u)
    for (int i = 0; i < D; ++i) acc += Wh[(size_t)i * D + tid] * su[i];
    float nv = sqrtf(block_sum(acc * acc, sred));
    sv[tid] = acc / (nv + SN_EPS);
    __syncthreads();
    float acc2 = 0.f;                                 // u = normalize(W v)
    for (int j = 0; j < D; ++j) acc2 += Wh[(size_t)tid * D + j] * sv[j];
    float nu = sqrtf(block_sum(acc2 * acc2, sred));
    su[tid] = acc2 / (nu + SN_EPS);
    __syncthreads();
  }

  float w = 0.f;                                      // sigma = |u . (W v)|
  for (int j = 0; j < D; ++j) w += Wh[(size_t)tid * D + j] * sv[j];
  float sigma = block_sum(su[tid] * w, sred);

  if (tid == 0) {
    ws_scale[0] = TARGET_RADIUS / (fabsf(sigma) + SN_EPS);
    *ctr = 0u;
  }
}

// -------------------------------------------------------------------------
// Kernel 2: transposed weight buffers (scale folded into WhT), h0 copy.
// -------------------------------------------------------------------------
__global__ void k_prep(const float* __restrict__ Wx,
                       const float* __restrict__ Wh,
                       const float* __restrict__ h0,
                       const float* __restrict__ ws_scale,
                       float* __restrict__ WxT,
                       float* __restrict__ WhT,
                       float* __restrict__ hAll,
                       int D, int B) {
  const int e = blockIdx.x;
  const float s = ws_scale[0];
  for (int k = threadIdx.x; k < D; k += blockDim.x) {
    WxT[(size_t)k * D + e] = Wx[(size_t)e * D + k];
    WhT[(size_t)k * D + e] = Wh[(size_t)e * D + k] * s;
  }
  if ((int)blockIdx.x < B) {
    const size_t row = (size_t)blockIdx.x * D;
    for (int j = threadIdx.x; j < D; j += blockDim.x)
      hAll[row + j] = h0[row + j];
  }
}

// -------------------------------------------------------------------------
// Kernel 3: pre = x @ Wx^T + b  ->  h_all slots [1..T].  f32 WMMA 16x16x4.
// Memory-bound (x 64MB in, pre 64MB out); L2 (192MB) holds the reuse, so
// operands stream straight to registers with running-pointer addressing.
// -------------------------------------------------------------------------
__global__ void __launch_bounds__(256)
k_pregemm(const float* __restrict__ x,
          const float* __restrict__ WxT,
          const float* __restrict__ bias,
          float* __restrict__ hAll,
          int D, int B) {
  const int lane  = threadIdx.x & 31;
  const int wave  = threadIdx.x >> 5;
  const int mbase = blockIdx.x * 16;
  const int nbase = blockIdx.y * 128 + wave * 16;
  const int nl    = lane & 15;
  const int kofs  = (lane >> 4) * 2;

  const float* ap = x   + (size_t)(mbase + nl) * D + kofs;   // A[m, k..k+1]
  const float* bp = WxT + (size_t)kofs * D + nbase + nl;     // B[k, n]
  __builtin_prefetch(ap, 0, 3);                              // global_prefetch_b8

  v8f acc = {};
#pragma unroll 4
  for (int k = 0; k < D; k += 4) {
    v2f a = *(const v2f*)ap;
    v2f b;
    b.x = bp[0];
    b.y = bp[D];
    acc = __builtin_amdgcn_wmma_f32_16x16x4_f32(
        false, a, false, b, (short)0, acc, false, false);
    ap += 4;
    bp += (size_t)4 * D;
  }

  const float bval = bias[nbase + nl];
#pragma unroll
  for (int r = 0; r < 8; ++r) {
    const int m = (lane < 16) ? r : (r + 8);
    hAll[(size_t)(mbase + m + B) * D + nbase + nl] = acc[r] + bval;
  }
}

// -------------------------------------------------------------------------
// Kernel 4: recurrence h_t = pre_t + h_{t-1} @ Wh^T, fused h^2*sigmoid(h).
// 32 blocks x 2 waves; block owns 32 N-columns. TDM stages the block's
// Wh^T slice (128KB) into LDS once and h_{t-1} (32KB) each step; inner
// loop is pure ds_load + v_wmma. Grid barrier: agent-scope rel/acq atomic.
// Dynamic LDS: Bs [D x 32] + hsh [16 x D] (rows B..15 zeroed once so the
// padded-M A fragment needs no per-iteration select).
// -------------------------------------------------------------------------
__global__ void __launch_bounds__(64)
k_recur(const float* __restrict__ WhT,
        float* __restrict__ hAll,
        float* __restrict__ out,
        unsigned* __restrict__ ctr,
        int D, int B, int T, int NB) {
  extern __shared__ float smem[];
  float* Bs  = smem;                   // [D][32]  Wh^T slice, row pitch 32
  float* hsh = smem + (size_t)32 * D;  // [16][D] zero-padded h staging

  const int lane  = threadIdx.x & 31;
  const int wave  = threadIdx.x >> 5;
  const int nl    = lane & 15;
  const int kofs  = (lane >> 4) * 2;
  const int mrow  = lane & 15;
  const int nbase = blockIdx.x * 32 + wave * 16;

  // Stage this block's Wh^T column slice via the Tensor Data Mover (once).
  if (wave == 0) {
    tdm_load_2d_f32((unsigned)(size_t)Bs, WhT + (size_t)blockIdx.x * 32,
                    /*tile_x=*/32, /*tile_y=*/(unsigned)D,
                    /*tensor_x=*/(unsigned)D, /*tensor_y=*/(unsigned)D,
                    /*stride_x=*/(unsigned)D);
  }
  // Zero padded A rows (B..15) once.
  for (int i = B * D + threadIdx.x; i < 16 * D; i += blockDim.x) hsh[i] = 0.f;
  __syncthreads();

  for (int t = 0; t < T; ++t) {
    // DMA h_{t-1} (contiguous B*D f32) into LDS; wait TENSORcnt.
    if (wave == 0) {
      tdm_load_2d_f32((unsigned)(size_t)hsh, hAll + (size_t)t * B * D,
                      /*tile_x=*/(unsigned)(B * D), /*tile_y=*/1u,
                      /*tensor_x=*/(unsigned)(B * D), /*tensor_y=*/1u,
                      /*stride_x=*/(unsigned)(B * D));
      __builtin_amdgcn_s_wait_tensorcnt(0);
    }
    __syncthreads();

    const float* ap = hsh + (size_t)mrow * D + kofs;            // A from LDS
    const float* bp = Bs + (size_t)kofs * 32 + wave * 16 + nl;  // B from LDS
    v8f acc = {};
#pragma unroll 8
    for (int k = 0; k < D; k += 4) {
      v2f a = *(const v2f*)ap;                             // ds_load_b64
      v2f b;
      b.x = bp[0];                                         // row k+kofs
      b.y = bp[32];                                        // row k+kofs+1
      acc = __builtin_amdgcn_wmma_f32_16x16x4_f32(
          false, a, false, b, (short)0, acc, false, false);
      ap += 4;
      bp += 128;                                           // 4 rows * pitch 32
    }

    float* hnew = hAll + (size_t)(t + 1) * B * D;          // holds pre_t
#pragma unroll
    for (int r = 0; r < 8; ++r) {
      const int m = (lane < 16) ? r : (r + 8);
      if (m < B) {
        const size_t idx = (size_t)m * D + nbase + nl;
        const float h = hnew[idx] + acc[r];
        hnew[idx] = h;
        const float sg = 1.0f / (1.0f + __expf(-h));
        out[(size_t)t * B * D + idx] = h * h * sg;         // h * silu(h)
      }
    }

    // ---- grid-wide barrier (release/acquire, agent scope) ----
    __syncthreads();
    if (threadIdx.x == 0) {
      __hip_atomic_fetch_add(ctr, 1u, __ATOMIC_RELEASE,
                             __HIP_MEMORY_SCOPE_AGENT);
      const unsigned target = (unsigned)(NB * (t + 1));
      while (__hip_atomic_load(ctr, __ATOMIC_ACQUIRE,
                               __HIP_MEMORY_SCOPE_AGENT) < target) {
        __builtin_amdgcn_s_sleep(2);
      }
    }
    __syncthreads();
  }
}

// -------------------------------------------------------------------------
// Host launcher
// -------------------------------------------------------------------------
extern "C" void kernel_launch(void* const* d_in, const int* in_sizes, int n_in,
                              void* d_out, int out_size, void* d_ws, size_t ws_size,
                              hipStream_t stream) {
  const float* x  = (const float*)d_in[0];
  const float* h0 = (const float*)d_in[1];
  const float* Wx = (const float*)d_in[2];
  const float* Wh = (const float*)d_in[3];
  const float* b  = (const float*)d_in[4];
  const float* u  = (const float*)d_in[5];

  const int D = in_sizes[4];                 // bias length
  const int B = in_sizes[1] / D;             // h0 = [B, D]
  const int T = in_sizes[0] / (B * D);       // x  = [T, B, D]

  float* out  = (float*)d_out;               // output  [T, B, D]
  float* hAll = out + (size_t)T * B * D;     // h_all   [T+1, B, D]

  float*    wsf   = (float*)d_ws;
  float*    scale = wsf;                     // spectral scale
  unsigned* ctr   = (unsigned*)(wsf + 1);    // grid barrier counter
  float*    WxT   = wsf + 256;               // D*D transposed Wx
  float*    WhT   = WxT + (size_t)D * D;     // D*D transposed+scaled Wh

  k_spectral<<<1, D, 0, stream>>>(Wh, u, scale, ctr, D);
  k_prep<<<D, 256, 0, stream>>>(Wx, Wh, h0, scale, WxT, WhT, hAll, D, B);

  dim3 g3((T * B) / 16, D / 128);
  k_pregemm<<<g3, 256, 0, stream>>>(x, WxT, b, hAll, D, B);

  const int NB = D / 32;                     // 32 blocks x 2 waves x 16 cols
  const size_t lds = (size_t)(32 * D + 16 * D) * sizeof(float);  // 192KB
  k_recur<<<NB, 64, lds, stream>>>(WhT, hAll, out, ctr, D, B, T, NB);
}